// A2C_Trading_17377437679802
// MI455X (gfx1250) — compile-verified
//
#include <hip/hip_runtime.h>
#include <hip/hip_bf16.h>

// Problem constants (match reference)
#define TT   65536
#define DIN  128
#define HH   64
#define G4H  256   // 4*H
#define AA   8

typedef float v2f __attribute__((ext_vector_type(2)));
typedef float v8f __attribute__((ext_vector_type(8)));

// ---------------------------------------------------------------------------
// fp32 WMMA GEMM:  G[chain][t][0:256] = X[t,:] @ W.T + (b_ih + b_hh)
// X: [T, K] row-major, W: [256, K] row-major (so B[k][n] = W[n*K+k]).
// One wave per 16x16 C tile, V_WMMA_F32_16X16X4_F32, K/4 k-steps.
// grid = (T/64, 16, 2), block = 128 (4 waves; wave w -> tileM sub-block).
// ---------------------------------------------------------------------------
__global__ __launch_bounds__(128)
void wmma_gemm_ih(const float* __restrict__ X0, const float* __restrict__ X1, int K,
                  const float* __restrict__ W0, const float* __restrict__ bi0, const float* __restrict__ bh0,
                  const float* __restrict__ W1, const float* __restrict__ bi1, const float* __restrict__ bh1,
                  float* __restrict__ G)
{
    const int chain = blockIdx.z;
    const float* __restrict__ X  = chain ? X1 : X0;
    const float* __restrict__ W  = chain ? W1 : W0;
    const float* __restrict__ bi = chain ? bi1 : bi0;
    const float* __restrict__ bh = chain ? bh1 : bh0;

    const int wave  = threadIdx.x >> 5;
    const int lane  = threadIdx.x & 31;
    const int half  = lane >> 4;       // 0: lanes 0-15, 1: lanes 16-31
    const int l     = lane & 15;

    const int tileM = (blockIdx.x * 4 + wave) * 16;
    const int tileN = blockIdx.y * 16;

    // C init: every VGPR of a lane maps to column (tileN + l) -> broadcast bias.
    const int col = tileN + l;
    const float bias = bi[col] + bh[col];
    v8f c;
    #pragma unroll
    for (int v = 0; v < 8; ++v) c[v] = bias;

    const int rowA = tileM + l;   // A fragment row for this lane
    const int rowB = tileN + l;   // W row supplying B column

    for (int k = 0; k < K; k += 4) {
        const int kk = k + 2 * half;
        // A: 16x4 f32 fragment (lane<16 -> K=k,k+1 ; lane>=16 -> K=k+2,k+3)
        float2 av = *reinterpret_cast<const float2*>(X + (size_t)rowA * K + kk);
        // B: 4x16 f32 fragment; B[k][n] = W[n][k] -> contiguous pair in W's row
        float2 bv = *reinterpret_cast<const float2*>(W + (size_t)rowB * K + kk);
        v2f a; a[0] = av.x; a[1] = av.y;
        v2f b; b[0] = bv.x; b[1] = bv.y;
        c = __builtin_amdgcn_wmma_f32_16x16x4_f32(false, a, false, b, (short)0, c,
                                                  false, false);
    }

    // D/C layout: lane<16 -> M = v, lane>=16 -> M = v+8 ; N = lane&15
    float* __restrict__ Gc = G + (size_t)chain * TT * G4H;
    #pragma unroll
    for (int v = 0; v < 8; ++v) {
        const int row = tileM + v + 8 * half;
        Gc[(size_t)row * G4H + col] = c[v];
    }
}

// ---------------------------------------------------------------------------
// Sequential LSTM scan for one layer. grid = 2 (chain), block = 256.
// Thread j owns gate j: w_hh[j][0:64] resident in VGPRs; h broadcast via LDS.
// G[t] register-prefetched 4 steps deep to hide global latency.
// ---------------------------------------------------------------------------
__device__ __forceinline__ float fsig(float x)  { return 1.0f / (1.0f + __expf(-x)); }
__device__ __forceinline__ float ftanh(float x) { float e = __expf(-2.0f * x); return (1.0f - e) / (1.0f + e); }

__global__ __launch_bounds__(256)
void lstm_scan(const float* __restrict__ G,        // [2][T][256] preactivations (input part + bias)
               const float* __restrict__ Whh0,     // chain0 w_hh [256][64]
               const float* __restrict__ Whh1,     // chain1 w_hh [256][64]
               float* __restrict__ Y,              // [2][T][64] layer outputs (nullptr for layer 1)
               float* __restrict__ out,            // d_out (521 floats)
               int layer,
               const float* __restrict__ aW, const float* __restrict__ ab,
               const float* __restrict__ cW, const float* __restrict__ cb)
{
    const int chain = blockIdx.x;
    const int j     = threadIdx.x;            // gate index 0..255

    const float* __restrict__ Gc  = G + (size_t)chain * TT * G4H;
    const float* __restrict__ Whh = chain ? Whh1 : Whh0;
    float*       __restrict__ Yc  = Y ? (Y + (size_t)chain * TT * HH) : nullptr;

    __shared__ __align__(16) float hbuf[2][HH];
    __shared__ float gact[G4H];

    if (j < HH) { hbuf[0][j] = 0.0f; hbuf[1][j] = 0.0f; }
    __syncthreads();

    // Recurrent weights resident in VGPRs (64 floats / thread).
    float4 w4[16];
    const float4* __restrict__ Wrow = reinterpret_cast<const float4*>(Whh + (size_t)j * HH);
    #pragma unroll
    for (int q = 0; q < 16; ++q) w4[q] = Wrow[q];

    const int type = j >> 6;                  // 0:i 1:f 2:g 3:o
    float cstate = 0.0f;                      // valid in threads j<64

    // 4-deep prefetch of gate preactivations.
    float gpre[4];
    #pragma unroll
    for (int p = 0; p < 4; ++p) gpre[p] = Gc[(size_t)p * G4H + j];

    for (int t = 0; t < TT; ++t) {
        const float gcur = gpre[t & 3];
        if (t + 4 < TT) gpre[t & 3] = Gc[(size_t)(t + 4) * G4H + j];

        // gate_j = gcur + dot(w_hh[j,:], h)   (h via LDS b128 broadcasts)
        const float4* __restrict__ hb = reinterpret_cast<const float4*>(hbuf[t & 1]);
        float a0 = gcur, a1 = 0.0f, a2 = 0.0f, a3 = 0.0f;
        #pragma unroll
        for (int q = 0; q < 16; q += 4) {
            float4 h0 = hb[q + 0], h1 = hb[q + 1], h2 = hb[q + 2], h3 = hb[q + 3];
            a0 += w4[q + 0].x * h0.x + w4[q + 0].y * h0.y + w4[q + 0].z * h0.z + w4[q + 0].w * h0.w;
            a1 += w4[q + 1].x * h1.x + w4[q + 1].y * h1.y + w4[q + 1].z * h1.z + w4[q + 1].w * h1.w;
            a2 += w4[q + 2].x * h2.x + w4[q + 2].y * h2.y + w4[q + 2].z * h2.z + w4[q + 2].w * h2.w;
            a3 += w4[q + 3].x * h3.x + w4[q + 3].y * h3.y + w4[q + 3].z * h3.z + w4[q + 3].w * h3.w;
        }
        const float pre = (a0 + a1) + (a2 + a3);
        gact[j] = (type == 2) ? ftanh(pre) : fsig(pre);
        __syncthreads();

        if (j < HH) {
            const float iv = gact[j];
            const float fv = gact[HH + j];
            const float gv = gact[2 * HH + j];
            const float ov = gact[3 * HH + j];
            cstate = fv * cstate + iv * gv;
            const float hn = ov * ftanh(cstate);
            hbuf[(t + 1) & 1][j] = hn;
            if (Yc) Yc[(size_t)t * HH + j] = hn;
        }
        __syncthreads();
    }

    // d_out layout: [0,8) action_mean | [8] value | [9,137) ha | [137,265) ca
    //               | [265,393) hc | [393,521) cc
    const int base_h = chain ? 265 : 9;
    const int base_c = chain ? 393 : 137;
    const float* __restrict__ hf = hbuf[TT & 1];
    if (j < HH) {
        out[base_h + layer * HH + j] = hf[j];
        out[base_c + layer * HH + j] = cstate;
    }

    if (layer == 1) {
        if (chain == 0 && j < AA) {
            float s = ab[j];
            #pragma unroll 8
            for (int k = 0; k < HH; ++k) s += aW[j * HH + k] * hf[k];
            out[j] = ftanh(s) * 0.3f;
        } else if (chain == 1 && j == 0) {
            float s = cb[0];
            #pragma unroll 8
            for (int k = 0; k < HH; ++k) s += cW[k] * hf[k];
            out[AA] = s;
        }
    }
}

// ---------------------------------------------------------------------------
// Host launcher
// ---------------------------------------------------------------------------
extern "C" void kernel_launch(void* const* d_in, const int* in_sizes, int n_in,
                              void* d_out, int out_size, void* d_ws, size_t ws_size,
                              hipStream_t stream)
{
    (void)in_sizes; (void)n_in; (void)out_size; (void)ws_size;

    const float* state   = (const float*)d_in[0];
    // actor
    const float* a_wih0  = (const float*)d_in[1];
    const float* a_whh0  = (const float*)d_in[2];
    const float* a_bih0  = (const float*)d_in[3];
    const float* a_bhh0  = (const float*)d_in[4];
    const float* a_wih1  = (const float*)d_in[5];
    const float* a_whh1  = (const float*)d_in[6];
    const float* a_bih1  = (const float*)d_in[7];
    const float* a_bhh1  = (const float*)d_in[8];
    // critic
    const float* c_wih0  = (const float*)d_in[9];
    const float* c_whh0  = (const float*)d_in[10];
    const float* c_bih0  = (const float*)d_in[11];
    const float* c_bhh0  = (const float*)d_in[12];
    const float* c_wih1  = (const float*)d_in[13];
    const float* c_whh1  = (const float*)d_in[14];
    const float* c_bih1  = (const float*)d_in[15];
    const float* c_bhh1  = (const float*)d_in[16];
    // heads
    const float* aW = (const float*)d_in[17];
    const float* ab = (const float*)d_in[18];
    const float* cW = (const float*)d_in[19];
    const float* cb = (const float*)d_in[20];

    float* out = (float*)d_out;
    float* gx  = (float*)d_ws;                       // [2][T][256]  (128 MB, reused layer0/1)
    float* y0  = gx + (size_t)2 * TT * G4H;          // [2][T][64]   (32 MB)

    // Phase 1: layer-0 input GEMM (K = 128), bias folded into accumulator.
    wmma_gemm_ih<<<dim3(TT / 64, G4H / 16, 2), 128, 0, stream>>>(
        state, state, DIN,
        a_wih0, a_bih0, a_bhh0,
        c_wih0, c_bih0, c_bhh0, gx);

    // Phase 2: layer-0 recurrence; emits y0 and layer-0 final h/c.
    lstm_scan<<<2, 256, 0, stream>>>(gx, a_whh0, c_whh0, y0, out, 0,
                                     aW, ab, cW, cb);

    // Phase 3: layer-1 input GEMM (K = 64) from y0 (per-chain X).
    wmma_gemm_ih<<<dim3(TT / 64, G4H / 16, 2), 128, 0, stream>>>(
        y0, y0 + (size_t)TT * HH, HH,
        a_wih1, a_bih1, a_bhh1,
        c_wih1, c_bih1, c_bhh1, gx);

    // Phase 4: layer-1 recurrence + heads (action_mean, state_value).
    lstm_scan<<<2, 256, 0, stream>>>(gx, a_whh1, c_whh1, nullptr, out, 1,
                                     aW, ab, cW, cb);
}